// GCNLayers_15607911154176
// MI455X (gfx1250) — compile-verified
//
#include <hip/hip_runtime.h>
#include <cstdint>
#include <cstddef>

#define N_NODES 100000
#define N_EDGES 1600000
#define DIM 128
#define BN_EPS 1e-5f

typedef float v2f __attribute__((ext_vector_type(2)));
typedef float v8f __attribute__((ext_vector_type(8)));

// ---------------- degree / normalization ----------------
__global__ void deg_accum_kernel(const long long* __restrict__ dst,
                                 float* __restrict__ deg) {
    int e = blockIdx.x * blockDim.x + threadIdx.x;
    if (e < N_EDGES) atomicAdd(&deg[(int)dst[e]], 1.0f);
}

__global__ void dinv_kernel(float* __restrict__ deg) {
    int i = blockIdx.x * blockDim.x + threadIdx.x;
    if (i < N_NODES) deg[i] = rsqrtf(deg[i] + 1.0f);
}

// ---------------- GEMM: hw = act(h) @ W  (fp32 WMMA 16x16x4) ----------------
// Block = 256 threads = 8 waves. Each wave owns one 16-col tile and computes
// TWO 16-row tiles (block covers 32 rows), reusing B fragments held in
// registers for all of K. HAS_BN is a compile-time flag: BN+ReLU of the
// previous layer is applied to the A operand on the fly (branch-free loop).
template <bool HAS_BN>
__global__ void __launch_bounds__(256)
gemm_wmma_kernel(const float* __restrict__ H, const float* __restrict__ W,
                 const float* __restrict__ scale, const float* __restrict__ shift,
                 float* __restrict__ out) {
    const int lane = threadIdx.x & 31;
    const int wave = threadIdx.x >> 5;      // 0..7 -> 16-col tile
    const int l15  = lane & 15;
    const int hi   = lane >> 4;             // 0: K={0,1}, 1: K={2,3}
    const int koff = hi * 2;
    const int col0 = wave * 16;
    const int row0 = blockIdx.x * 32;

    // Preload the full-K B fragments for this wave's column tile (64 VGPRs).
    v2f b[32];
    #pragma unroll
    for (int i = 0; i < 32; ++i) {
        const int k = i * 4 + koff;
        b[i].x = W[(size_t)k       * DIM + col0 + l15];
        b[i].y = W[(size_t)(k + 1) * DIM + col0 + l15];
    }

    const float* hrow0 = H + (size_t)(row0 + l15) * DIM;
    const float* hrow1 = hrow0 + (size_t)16 * DIM;

    v8f c0 = {};
    v8f c1 = {};
    #pragma unroll    // full unroll: b[i] indices must be compile-time constants
    for (int i = 0; i < 32; ++i) {
        const int k = i * 4 + koff;
        float a00 = hrow0[k];
        float a01 = hrow0[k + 1];
        float a10 = hrow1[k];
        float a11 = hrow1[k + 1];
        if (HAS_BN) {  // compile-time branch: no per-iteration scc checks
            const float s0 = scale[k],     t0 = shift[k];
            const float s1 = scale[k + 1], t1 = shift[k + 1];
            a00 = fmaxf(0.0f, a00 * s0 + t0);
            a01 = fmaxf(0.0f, a01 * s1 + t1);
            a10 = fmaxf(0.0f, a10 * s0 + t0);
            a11 = fmaxf(0.0f, a11 * s1 + t1);
        }
        v2f a0; a0.x = a00; a0.y = a01;
        v2f a1; a1.x = a10; a1.y = a11;
        c0 = __builtin_amdgcn_wmma_f32_16x16x4_f32(
                 false, a0, false, b[i], (short)0, c0, false, false);
        c1 = __builtin_amdgcn_wmma_f32_16x16x4_f32(
                 false, a1, false, b[i], (short)0, c1, false, false);
    }

    // D layout: M = vgpr + 8*(lane>=16), N = lane%16
    float* o = out + (size_t)(col0 + l15);
    const int m0 = row0 + hi * 8;
    #pragma unroll
    for (int v = 0; v < 8; ++v) {
        o[(size_t)(m0 + v)      * DIM] = c0[v];
        o[(size_t)(m0 + 16 + v) * DIM] = c1[v];
    }
}

// ---------------- agg = hw * dinv^2 + bias (self-loop + bias init) ----------------
__global__ void self_init_kernel(const float* __restrict__ hw,
                                 const float* __restrict__ dinv,
                                 const float* __restrict__ bias,
                                 float* __restrict__ agg) {
    int idx = blockIdx.x * blockDim.x + threadIdx.x;   // one float4 per thread
    if (idx >= N_NODES * (DIM / 4)) return;
    const int row = idx >> 5;
    const int f4  = idx & 31;
    const float d  = dinv[row];
    const float d2 = d * d;
    const float4 h4 = ((const float4*)hw)[idx];
    const float4 b4 = ((const float4*)bias)[f4];
    float4 r;
    r.x = h4.x * d2 + b4.x;
    r.y = h4.y * d2 + b4.y;
    r.z = h4.z * d2 + b4.z;
    r.w = h4.w * d2 + b4.w;
    ((float4*)agg)[idx] = r;
}

// ---------------- edge scatter: agg[dst] += hw[src] * norm ----------------
// One wave per edge: 32 lanes x float4 = 128 features. hw/agg tables are
// 51.2 MB each -> resident in the 192 MB L2, so atomics stay on-chip.
__global__ void __launch_bounds__(256)
scatter_kernel(const long long* __restrict__ src, const long long* __restrict__ dst,
               const float* __restrict__ dinv, const float* __restrict__ hw,
               float* __restrict__ agg) {
    const int lane = threadIdx.x & 31;
    const long long e = (long long)blockIdx.x * 8 + (threadIdx.x >> 5);
    if (e >= N_EDGES) return;
    const int s = (int)src[e];
    const int d = (int)dst[e];
    const float norm = dinv[s] * dinv[d];
    const float4 v = ((const float4*)(hw + (size_t)s * DIM))[lane];
    float* out = agg + (size_t)d * DIM + lane * 4;
    atomicAdd(out + 0, v.x * norm);
    atomicAdd(out + 1, v.y * norm);
    atomicAdd(out + 2, v.z * norm);
    atomicAdd(out + 3, v.w * norm);
}

// ---------------- BN statistics: per-feature sum / sumsq ----------------
__global__ void bn_reduce_kernel(const float* __restrict__ h, float* __restrict__ bn) {
    const int f  = threadIdx.x;                 // 0..127
    const int r0 = blockIdx.x * 250;            // 400 blocks x 250 rows = 100000
    float s = 0.0f, q = 0.0f;
    for (int r = r0; r < r0 + 250; ++r) {
        const float v = h[(size_t)r * DIM + f];
        s += v;
        q += v * v;
    }
    atomicAdd(&bn[f], s);
    atomicAdd(&bn[DIM + f], q);
}

__global__ void bn_finalize_kernel(float* __restrict__ bn,
                                   const float* __restrict__ gamma,
                                   const float* __restrict__ beta) {
    const int f = threadIdx.x;
    const float inv_n = 1.0f / (float)N_NODES;
    const float mean = bn[f] * inv_n;
    const float var  = bn[DIM + f] * inv_n - mean * mean;   // biased var
    const float sc   = gamma[f] * rsqrtf(var + BN_EPS);
    bn[2 * DIM + f] = sc;
    bn[3 * DIM + f] = beta[f] - mean * sc;
}

// ---------------- final BN + ReLU -> d_out ----------------
__global__ void bn_apply_relu_kernel(const float* __restrict__ agg,
                                     const float* __restrict__ bn,
                                     float* __restrict__ out) {
    int idx = blockIdx.x * blockDim.x + threadIdx.x;
    if (idx >= N_NODES * (DIM / 4)) return;
    const int f4 = idx & 31;
    const float4 v  = ((const float4*)agg)[idx];
    const float4 sc = ((const float4*)(bn + 2 * DIM))[f4];
    const float4 sh = ((const float4*)(bn + 3 * DIM))[f4];
    float4 r;
    r.x = fmaxf(0.0f, v.x * sc.x + sh.x);
    r.y = fmaxf(0.0f, v.y * sc.y + sh.y);
    r.z = fmaxf(0.0f, v.z * sc.z + sh.z);
    r.w = fmaxf(0.0f, v.w * sc.w + sh.w);
    ((float4*)out)[idx] = r;
}

extern "C" void kernel_launch(void* const* d_in, const int* in_sizes, int n_in,
                              void* d_out, int out_size, void* d_ws, size_t ws_size,
                              hipStream_t stream) {
    const float* x = (const float*)d_in[0];
    const long long* ei   = (const long long*)d_in[1];
    const long long* esrc = ei;              // edge_index[0]
    const long long* edst = ei + N_EDGES;    // edge_index[1]

    // workspace layout
    float* dinv = (float*)d_ws;                        // N_NODES floats
    float* bn   = dinv + N_NODES;                      // 4*DIM: sum, sq, scale, shift
    float* bufA = bn + 4 * DIM;                        // N*DIM (hw)
    float* bufB = bufA + (size_t)N_NODES * DIM;        // N*DIM (agg)

    // degree + dinv_sqrt (recomputed every call: deterministic)
    hipMemsetAsync(dinv, 0, N_NODES * sizeof(float), stream);
    deg_accum_kernel<<<(N_EDGES + 255) / 256, 256, 0, stream>>>(edst, dinv);
    dinv_kernel<<<(N_NODES + 255) / 256, 256, 0, stream>>>(dinv);

    const int gemm_grid = N_NODES / 32;                      // 3125 (exact)
    const int elem_grid = (N_NODES * (DIM / 4) + 255) / 256; // 12500
    const int scat_grid = N_EDGES / 8;                       // 200000 (exact)

    const float* hin = x;
    for (int l = 0; l < 3; ++l) {
        const float* W     = (const float*)d_in[2 + 4 * l];
        const float* b     = (const float*)d_in[3 + 4 * l];
        const float* gamma = (const float*)d_in[4 + 4 * l];
        const float* beta  = (const float*)d_in[5 + 4 * l];

        // hw = act(hin) @ W  (WMMA; BN+ReLU of previous layer fused into A-load)
        if (l == 0) {
            gemm_wmma_kernel<false><<<gemm_grid, 256, 0, stream>>>(
                hin, W, nullptr, nullptr, bufA);
        } else {
            gemm_wmma_kernel<true><<<gemm_grid, 256, 0, stream>>>(
                hin, W, bn + 2 * DIM, bn + 3 * DIM, bufA);
        }
        // agg = hw*dinv^2 + bias, then scatter edges
        self_init_kernel<<<elem_grid, 256, 0, stream>>>(bufA, dinv, b, bufB);
        scatter_kernel<<<scat_grid, 256, 0, stream>>>(esrc, edst, dinv, bufA, bufB);
        // BN stats for this layer's output
        hipMemsetAsync(bn, 0, 2 * DIM * sizeof(float), stream);
        bn_reduce_kernel<<<N_NODES / 250, DIM, 0, stream>>>(bufB, bn);
        bn_finalize_kernel<<<1, DIM, 0, stream>>>(bn, gamma, beta);

        hin = bufB;  // next GEMM reads agg with fused BN+ReLU
    }

    // final layer: materialize BN+ReLU into d_out
    bn_apply_relu_kernel<<<elem_grid, 256, 0, stream>>>(bufB, bn, (float*)d_out);
}